// KalmanFilter_49314814492675
// MI455X (gfx1250) — compile-verified
//
#include <hip/hip_runtime.h>

// CDNA5 / gfx1250. Persistent single-workgroup Kalman scan (latency-bound
// sequential recursion). All matmuls use V_WMMA_F32_16X16X4_F32 (full f32).
// Loop-invariant F / F^T / Q / H fragments live in registers across all
// 20000 steps; the per-step mask byte is software-pipelined (prefetched one
// iteration ahead) so no global-load wait sits in front of the WMMA chains.

typedef __attribute__((ext_vector_type(2))) float v2f;
typedef __attribute__((ext_vector_type(8))) float v8f;

#define LDF 33   // stride for 32-wide matrices (odd -> no LDS bank conflicts)
#define LDP 17   // stride for 16-wide matrices

// ---- WMMA f32 16x16x4 fragment helpers (ISA 7.12.2 layouts) ----
// A (16x4): lane l, reg r -> A[l%16][r + 2*(l/16)]
__device__ __forceinline__ v2f frag_a(const float* M, int ld, int row0, int k0, int lane) {
    int m  = lane & 15;
    int kh = (lane >> 4) << 1;
    const float* p = &M[(row0 + m) * ld + k0 + kh];
    v2f a; a.x = p[0]; a.y = p[1];
    return a;
}
// B (4x16) row-major source: lane l, reg r -> B[r + 2*(l/16)][l%16]
__device__ __forceinline__ v2f frag_b(const float* M, int ld, int col0, int k0, int lane) {
    int n  = lane & 15;
    int kh = (lane >> 4) << 1;
    v2f b;
    b.x = M[(k0 + kh)     * ld + col0 + n];
    b.y = M[(k0 + kh + 1) * ld + col0 + n];
    return b;
}
// B = Mt^T where Mt is row-major: B[k][n] = Mt[n][k]
__device__ __forceinline__ v2f frag_bt(const float* Mt, int ld, int col0, int k0, int lane) {
    int n  = lane & 15;
    int kh = (lane >> 4) << 1;
    const float* p = &Mt[(col0 + n) * ld + k0 + kh];
    v2f b; b.x = p[0]; b.y = p[1];
    return b;
}
// C/D (16x16 f32): lane l, reg r -> D[r + 8*(l/16)][l%16]
__device__ __forceinline__ v8f frag_c(const float* M, int ld, int row0, int col0, int lane) {
    int n  = lane & 15;
    int mh = (lane >> 4) << 3;
    v8f c;
#pragma unroll
    for (int r = 0; r < 8; ++r) c[r] = M[(row0 + mh + r) * ld + col0 + n];
    return c;
}
__device__ __forceinline__ void frag_st(float* M, int ld, int row0, int col0, int lane, v8f d) {
    int n  = lane & 15;
    int mh = (lane >> 4) << 3;
#pragma unroll
    for (int r = 0; r < 8; ++r) M[(row0 + mh + r) * ld + col0 + n] = d[r];
}

#define WMMA_F32(a, b, c) \
    __builtin_amdgcn_wmma_f32_16x16x4_f32(false, (a), false, (b), (short)0, (c), false, false)

__global__ __launch_bounds__(256) void kalman_scan(
    const float* __restrict__ obs, const unsigned char* __restrict__ mask,
    const float* __restrict__ F_g, const float* __restrict__ b_g,
    const float* __restrict__ H_g, const float* __restrict__ d_g,
    const float* __restrict__ Qraw, const float* __restrict__ Rraw,
    const float* __restrict__ m0, const float* __restrict__ P0raw,
    float* __restrict__ means, float* __restrict__ covs, int T)
{
    __shared__ float sF[32 * LDF];    // F (fragment source + m update)
    __shared__ float sQ[32 * LDF];    // Q = tril(Qraw) tril(Qraw)^T (fragment source)
    __shared__ float sH[16 * LDF];    // H (fragment source + innovation dot)
    __shared__ float sR[16 * LDP];    // R = tril(Rraw) tril(Rraw)^T
    __shared__ float sP[32 * LDF];    // P_pred (recursive state)
    __shared__ float sPHt[32 * LDP];  // P @ Hm^T
    __shared__ float sS[16 * LDP];    // Rm, then S
    __shared__ float sSn[16 * LDP];   // -S^{-1}
    __shared__ float sKn[32 * LDP];   // -K = PHt @ (-S^{-1})
    __shared__ float sPf[32 * LDF];   // filtered covariance
    __shared__ float sFP[32 * LDF];   // F @ P_f
    __shared__ float sd[16], sb[32], sm[32], sv[16], sw[16], smf[32];

    const int tid  = threadIdx.x;
    const int lane = tid & 31;
    // force scalar (uniform) branching so WMMA always executes with EXEC all-1s
    const int wave = __builtin_amdgcn_readfirstlane(tid >> 5);

    // ---------------- one-time setup ----------------
    for (int idx = tid; idx < 32 * 32; idx += 256) {
        int i = idx >> 5, j = idx & 31;
        sF[i * LDF + j] = F_g[idx];
    }
    for (int idx = tid; idx < 16 * 32; idx += 256) {
        int i = idx >> 5, j = idx & 31;
        sH[i * LDF + j] = H_g[idx];
    }
    if (tid < 32) { sb[tid] = b_g[tid]; sm[tid] = m0[tid]; }
    if (tid < 16) sd[tid] = d_g[tid];
    // PosDef transforms: C = tril(raw) @ tril(raw)^T
    for (int idx = tid; idx < 1024; idx += 256) {
        int i = idx >> 5, j = idx & 31;
        int km = (i < j) ? i : j;
        float aq = 0.f, ap = 0.f;
        for (int k = 0; k <= km; ++k) {
            aq += Qraw[i * 32 + k] * Qraw[j * 32 + k];
            ap += P0raw[i * 32 + k] * P0raw[j * 32 + k];
        }
        sQ[i * LDF + j] = aq;
        sP[i * LDF + j] = ap;   // P0
    }
    {
        int i = tid >> 4, j = tid & 15;   // exactly 256 threads
        int km = (i < j) ? i : j;
        float ar = 0.f;
        for (int k = 0; k <= km; ++k) ar += Rraw[i * 16 + k] * Rraw[j * 16 + k];
        sR[i * LDP + j] = ar;
    }
    __syncthreads();

    // ---- hoist loop-invariant fragments into registers (per wave's tile) ----
    const int wv = wave & 3;
    const int tr0 = (wv >> 1) * 16;   // output-tile row for 32x32 products
    const int tc0 = (wv & 1)  * 16;   // output-tile col for 32x32 products
    v2f fA[8], fBt[8], hA[8], hBt[8];
#pragma unroll
    for (int k = 0; k < 32; k += 4) {
        fA[k >> 2]  = frag_a (sF, LDF, tr0, k, lane);  // F row-strip (phase G)
        fBt[k >> 2] = frag_bt(sF, LDF, tc0, k, lane);  // F^T col-strip (phase H)
        hA[k >> 2]  = frag_a (sH, LDF, 0,   k, lane);  // H rows (phase C, scale by mf)
        hBt[k >> 2] = frag_bt(sH, LDF, 0,   k, lane);  // H^T cols (phase B, scale by mf)
    }
    const v8f qC = frag_c(sQ, LDF, tr0, tc0, lane);    // Q tile (phase H accumulator)

    // Software-pipelined mask: each thread carries mf[lane&15] for step t.
    float mfl = (mask[lane & 15] != 0) ? 1.f : 0.f;

    // ---------------- sequential scan ----------------
    for (int t = 0; t < T; ++t) {
        const unsigned char* mkt = mask + (size_t)t * 16;   // jax bool: 1 byte/elem

        // Phase B: PHt = P @ (H^T diag(mf)) (waves 0,1); innovation v (wave 2);
        //          Rm -> sS (waves 3..7). Ordered vs phase C by the barrier below.
        if (wave < 2) {
            int r0 = wave * 16;
            v8f acc = {0.f, 0.f, 0.f, 0.f, 0.f, 0.f, 0.f, 0.f};
#pragma unroll
            for (int k = 0; k < 32; k += 4) {
                v2f a  = frag_a(sP, LDF, r0, k, lane);
                v2f bb = hBt[k >> 2];
                bb.x *= mfl; bb.y *= mfl;        // mask scale, no memory access
                acc = WMMA_F32(a, bb, acc);
            }
            frag_st(sPHt, LDP, r0, 0, lane, acc);
        } else if (wave == 2) {
            if (lane < 16) {
                float acc = obs[(size_t)t * 16 + lane] - sd[lane];
                for (int i = 0; i < 32; ++i) acc -= sH[lane * LDF + i] * sm[i];
                sv[lane] = mfl * acc;   // v = mf * (y - d - H m)  (mask folded)
            }
        } else {
            for (int idx = tid - 96; idx < 256; idx += 160) {  // waves 3..7: Rm
                int j = idx >> 4, i = idx & 15;
                float mj = (mkt[j] != 0) ? 1.f : 0.f;
                float mi = (mkt[i] != 0) ? 1.f : 0.f;
                sS[j * LDP + i] = sR[j * LDP + i] * mj * mi + ((i == j) ? (1.f - mj) : 0.f);
            }
        }
        __syncthreads();

        // Phase C (wave 0 only): S = (diag(mf) H) @ PHt + Rm, then in-wave
        // Gauss-Jordan inverse of SPD S via lane shuffles; store -S^{-1}.
        if (wave == 0) {
            v8f acc = frag_c(sS, LDP, 0, 0, lane);
#pragma unroll
            for (int k = 0; k < 32; k += 4) {
                v2f a = hA[k >> 2];
                a.x *= mfl; a.y *= mfl;
                v2f bb = frag_b(sPHt, LDP, 0, k, lane);
                acc = WMMA_F32(a, bb, acc);
            }
            frag_st(sS, LDP, 0, 0, lane, acc);   // same-wave DS ops are in-order

            float c[16];
            if (lane < 16) {
#pragma unroll
                for (int i = 0; i < 16; ++i) c[i] = sS[i * LDP + lane];
            } else {
#pragma unroll
                for (int i = 0; i < 16; ++i) c[i] = ((lane - 16) == i) ? 1.f : 0.f;
            }
#pragma unroll
            for (int k = 0; k < 16; ++k) {
                float piv = __shfl(c[k], k, 32);
                float inv = 1.0f / piv;
                c[k] *= inv;
#pragma unroll
                for (int i = 0; i < 16; ++i) {
                    if (i == k) continue;
                    float f = __shfl(c[i], k, 32);
                    c[i] -= f * c[k];
                }
            }
            if (lane >= 16) {
#pragma unroll
                for (int i = 0; i < 16; ++i) sSn[i * LDP + (lane - 16)] = -c[i];
            }
        }

        // Prefetch next step's mask byte; latency overlaps phases E..H + barriers.
        {
            int tn = (t + 1 < T) ? (t + 1) : t;
            mfl = (mask[(size_t)tn * 16 + (lane & 15)] != 0) ? 1.f : 0.f;
        }
        __syncthreads();

        // Phase E: Kn = PHt @ (-S^{-1}) (waves 0,1); w = (-S^{-1}) v (wave 2)
        if (wave < 2) {
            int r0 = wave * 16;
            v8f acc = {0.f, 0.f, 0.f, 0.f, 0.f, 0.f, 0.f, 0.f};
#pragma unroll
            for (int k = 0; k < 16; k += 4) {
                v2f a  = frag_a(sPHt, LDP, r0, k, lane);
                v2f bb = frag_b(sSn, LDP, 0, k, lane);
                acc = WMMA_F32(a, bb, acc);
            }
            frag_st(sKn, LDP, r0, 0, lane, acc);
        } else if (wave == 2) {
            if (lane < 16) {
                float acc = 0.f;
                for (int l = 0; l < 16; ++l) acc += sSn[lane * LDP + l] * sv[l];
                sw[lane] = acc;
            }
        }
        __syncthreads();

        // Phase F: P_f = Kn @ PHt^T + P_pred  (P symmetric => Hm P = PHt^T) (waves 0-3)
        //          m_f = m_pred - PHt @ w (wave 4) and write filtered mean
        if (wave < 4) {
            v8f acc = frag_c(sP, LDF, tr0, tc0, lane);
#pragma unroll
            for (int k = 0; k < 16; k += 4) {
                v2f a  = frag_a(sKn, LDP, tr0, k, lane);
                v2f bb = frag_bt(sPHt, LDP, tc0, k, lane);
                acc = WMMA_F32(a, bb, acc);
            }
            frag_st(sPf, LDF, tr0, tc0, lane, acc);
        } else if (wave == 4) {
            float acc = sm[lane];
            for (int l = 0; l < 16; ++l) acc -= sPHt[lane * LDP + l] * sw[l];
            smf[lane] = acc;
            means[(size_t)t * 32 + lane] = acc;
        }
        __syncthreads();

        // Phase G: FP = F @ P_f (waves 0-3, F from registers);
        //          m_pred = F m_f + b (wave 4);
        //          stream covs[t] = P_f to HBM (waves 5-7, 128-bit stores)
        if (wave < 4) {
            v8f acc = {0.f, 0.f, 0.f, 0.f, 0.f, 0.f, 0.f, 0.f};
#pragma unroll
            for (int k = 0; k < 32; k += 4) {
                v2f bb = frag_b(sPf, LDF, tc0, k, lane);
                acc = WMMA_F32(fA[k >> 2], bb, acc);
            }
            frag_st(sFP, LDF, tr0, tc0, lane, acc);
        } else if (wave == 4) {
            float acc = sb[lane];
            for (int l = 0; l < 32; ++l) acc += sF[lane * LDF + l] * smf[l];
            sm[lane] = acc;
        } else {
            int base = tid - 160;                     // 0..95 over waves 5-7
            float4* out4 = (float4*)(covs + (size_t)t * 1024);
            for (int q = base; q < 256; q += 96) {    // 256 float4 = 32x32 tile
                int row = q >> 3, col = (q & 7) << 2;
                const float* p = &sPf[row * LDF + col];
                out4[q] = make_float4(p[0], p[1], p[2], p[3]);
            }
        }
        __syncthreads();

        // Phase H: P_pred = FP @ F^T + Q (waves 0-3; F^T and Q from registers)
        if (wave < 4) {
            v8f acc = qC;
#pragma unroll
            for (int k = 0; k < 32; k += 4) {
                v2f a = frag_a(sFP, LDF, tr0, k, lane);
                acc = WMMA_F32(a, fBt[k >> 2], acc);
            }
            frag_st(sP, LDF, tr0, tc0, lane, acc);
        }
        __syncthreads();
    }
}

extern "C" void kernel_launch(void* const* d_in, const int* in_sizes, int n_in,
                              void* d_out, int out_size, void* d_ws, size_t ws_size,
                              hipStream_t stream) {
    const float*         obs   = (const float*)d_in[0];
    const unsigned char* mask  = (const unsigned char*)d_in[1];  // jax bool: 1 byte/elem
    const float*         F     = (const float*)d_in[2];
    const float*         b     = (const float*)d_in[3];
    const float*         H     = (const float*)d_in[4];
    const float*         d     = (const float*)d_in[5];
    const float*         Qraw  = (const float*)d_in[6];
    const float*         Rraw  = (const float*)d_in[7];
    const float*         m0    = (const float*)d_in[8];
    const float*         P0raw = (const float*)d_in[9];

    const int T = in_sizes[0] / 16;
    float* means = (float*)d_out;
    float* covs  = means + (size_t)T * 32;

    hipLaunchKernelGGL(kalman_scan, dim3(1), dim3(256), 0, stream,
                       obs, mask, F, b, H, d, Qraw, Rraw, m0, P0raw,
                       means, covs, T);
}